// Mamba_75505525063769
// MI455X (gfx1250) — compile-verified
//
#include <hip/hip_runtime.h>

// Problem constants (from reference)
#define B_   64
#define L_   65
#define D_   512
#define DI_  1024
#define NS_  16
#define DTR_ 32
#define LH_  33     // mid = L//2 + 1
#define NLAYERS_ 2

typedef __attribute__((ext_vector_type(2))) float v2f;
typedef __attribute__((ext_vector_type(4))) float v4f;
typedef __attribute__((ext_vector_type(8))) float v8f;

enum { ACT_NONE = 0, ACT_SILU = 1, ACT_SOFTPLUS = 2 };

// ---- CDNA5 async global->LDS path (ASYNCcnt), with portable fallback ------
#if defined(__has_builtin)
#if __has_builtin(__builtin_amdgcn_global_load_async_to_lds_b128) && \
    __has_builtin(__builtin_amdgcn_global_load_async_to_lds_b32)
#define HAVE_ASYNC_LDS 1
#endif
#endif

#ifndef HAVE_ASYNC_LDS
#define HAVE_ASYNC_LDS 0
#endif

#if HAVE_ASYNC_LDS
#if __has_builtin(__builtin_amdgcn_s_wait_asynccnt)
#define ASYNC_WAIT(n) __builtin_amdgcn_s_wait_asynccnt(n)
#else
#define ASYNC_WAIT(n) asm volatile("s_wait_asynccnt %0" ::"n"(n) : "memory")
#endif
// Builtin signatures (from clang diagnostics):
//   b128: (AS1 int4*, AS3 int4*, int imm_offset, int cpol)
//   b32 : (AS1 int*,  AS3 int*,  int imm_offset, int cpol)
typedef int v4i_t __attribute__((vector_size(16)));
typedef __attribute__((address_space(1))) v4i_t* g128_t;
typedef __attribute__((address_space(3))) v4i_t* l128_t;
typedef __attribute__((address_space(1))) int*   g32_t;
typedef __attribute__((address_space(3))) int*   l32_t;
#else
#define ASYNC_WAIT(n) ((void)0)
#endif

__device__ __forceinline__ float apply_act(float val, int act)
{
  if (act == ACT_SILU)          return val / (1.f + __expf(-val));
  else if (act == ACT_SOFTPLUS) return (val > 20.f) ? val : __logf(1.f + __expf(val));
  return val;
}

// Stage a 32(N) x 32(K) B panel into LDS (K-major per column, stride 36).
// transB:  panel element (c,k) = Bw[(col0+c)*ldb + k0+k]   (rows are columns)
// !transB: panel element (c,k) = Bw[(k0+k)*ldb + col0+c]   (transposed store)
__device__ __forceinline__ void stage_bT(const float* Bw, float* tile,
                                         int col0, int k0, int ldb, int t)
{
  int c = t >> 3, k = (t & 7) << 2;                 // 16B per thread
  const float* g = Bw + (long long)(col0 + c) * ldb + (k0 + k);
  float* l = tile + c * 36 + k;
#if HAVE_ASYNC_LDS
  __builtin_amdgcn_global_load_async_to_lds_b128((g128_t)g, (l128_t)l, 0, 0);
#else
  *(v4f*)l = *(const v4f*)g;
#endif
}

__device__ __forceinline__ void stage_bN(const float* Bw, float* tile,
                                         int col0, int k0, int ldb, int t)
{
  int k = t >> 3, c4 = (t & 7) << 2;                // 4 scattered b32 per thread
  const float* g = Bw + (long long)(k0 + k) * ldb + (col0 + c4);
#pragma unroll
  for (int j = 0; j < 4; ++j) {
#if HAVE_ASYNC_LDS
    __builtin_amdgcn_global_load_async_to_lds_b32((g32_t)(g + j),
                                                  (l32_t)(tile + (c4 + j) * 36 + k), 0, 0);
#else
    tile[(c4 + j) * 36 + k] = g[j];
#endif
  }
}

__device__ __forceinline__ void epilogue_store(
    float* C, const float* bias, const v8f& acc0, const v8f& acc1,
    int M, int N, int ldc, int row0, int lh, int bcol0, int bcol1,
    bool hasT1, int act)
{
#pragma unroll
  for (int v = 0; v < 8; ++v) {
    int r = row0 + lh * 8 + v;
    if (r < M) {
      if (bcol0 < N) {
        float val = acc0[v];
        if (bias) val += bias[bcol0];
        C[(long long)r * ldc + bcol0] = apply_act(val, act);
      }
      if (hasT1 && bcol1 < N) {
        float val = acc1[v];
        if (bias) val += bias[bcol1];
        C[(long long)r * ldc + bcol1] = apply_act(val, act);
      }
    }
  }
}

// ---------------------------------------------------------------------------
// Generic batched GEMM on the CDNA5 fp32 WMMA path:
//   C[M,N] = act( A[M,K] * op(Bw) + bias ), op = transpose if transB
// One wave computes a 16x32 C stripe (two 16x16 WMMA tiles sharing the A
// fragment); block = 8 waves (wave32) -> 128x32 per block.
// When K%32==0 && N%32==0 (all large GEMMs here), the 32x32 B panel is staged
// into LDS via async global->LDS loads, double-buffered on ASYNCcnt, and
// reused by all 8 waves. A-row indices are clamped (not guarded) so the WMMA
// loop is branch-free with EXEC all-ones; OOB rows are simply never stored.
// ---------------------------------------------------------------------------
__global__ __launch_bounds__(256) void wmma_gemm_f32(
    const float* __restrict__ A, const float* __restrict__ Bw,
    const float* __restrict__ bias, float* __restrict__ C,
    int M, int N, int K, int lda, int ldb, int ldc,
    long long sA, long long sB, long long sC,
    int transB, int accum, int act)
{
  __shared__ float tileB[2][32 * 36];   // double-buffered B panel, 9216 B

  A  += (long long)blockIdx.z * sA;
  Bw += (long long)blockIdx.z * sB;
  C  += (long long)blockIdx.z * sC;

  const int lane = threadIdx.x;            // 0..31
  const int wave = threadIdx.y;            // 0..7
  const int t256 = wave * 32 + lane;       // 0..255
  const int row0 = (blockIdx.y * 8 + wave) * 16;
  const int col0 = blockIdx.x * 32;

  const int lm = lane & 15;
  const int lh = lane >> 4;
  const int arow  = row0 + lm;
  const int bcol0 = col0 + lm;
  const int bcol1 = col0 + 16 + lm;

  // Block-uniform LDS-path eligibility (holds for every large GEMM here).
  const bool use_lds = ((K & 31) == 0) && ((N & 31) == 0) &&
                       ((ldb & 3) == 0) && ((lda & 1) == 0);

  if (use_lds) {
    v8f acc0 = {}, acc1 = {};
    if (accum) {
#pragma unroll
      for (int v = 0; v < 8; ++v) {
        int r = row0 + lh * 8 + v;
        if (r < M) {
          acc0[v] = C[(long long)r * ldc + bcol0];
          acc1[v] = C[(long long)r * ldc + bcol1];
        }
      }
    }
    // Clamp A row: OOB rows compute garbage into C rows that are never stored.
    const int arc = (arow < M) ? arow : (M - 1);
    const float* ap = A + (long long)arc * lda + lh * 2;
    const int nst = K >> 5;

    if (transB) {
      stage_bT(Bw, tileB[0], col0, 0, ldb, t256);
      for (int s = 0; s < nst; ++s) {
        const int cb = s & 1;
        if (s + 1 < nst) { stage_bT(Bw, tileB[cb ^ 1], col0, (s + 1) << 5, ldb, t256); ASYNC_WAIT(1); }
        else             { ASYNC_WAIT(0); }
        __syncthreads();
        const float* bt0 = tileB[cb] + lm * 36 + lh * 2;
        const float* bt1 = tileB[cb] + (lm + 16) * 36 + lh * 2;
#pragma unroll
        for (int kk = 0; kk < 32; kk += 4) {
          v2f a  = *(const v2f*)ap; ap += 4;
          v2f b0 = *(const v2f*)(bt0 + kk);
          v2f b1 = *(const v2f*)(bt1 + kk);
          acc0 = __builtin_amdgcn_wmma_f32_16x16x4_f32(false, a, false, b0, (short)0, acc0, false, false);
          acc1 = __builtin_amdgcn_wmma_f32_16x16x4_f32(false, a, false, b1, (short)0, acc1, false, false);
        }
        __syncthreads();
      }
    } else {
      stage_bN(Bw, tileB[0], col0, 0, ldb, t256);
      for (int s = 0; s < nst; ++s) {
        const int cb = s & 1;
        if (s + 1 < nst) { stage_bN(Bw, tileB[cb ^ 1], col0, (s + 1) << 5, ldb, t256); ASYNC_WAIT(4); }
        else             { ASYNC_WAIT(0); }
        __syncthreads();
        const float* bt0 = tileB[cb] + lm * 36 + lh * 2;
        const float* bt1 = tileB[cb] + (lm + 16) * 36 + lh * 2;
#pragma unroll
        for (int kk = 0; kk < 32; kk += 4) {
          v2f a  = *(const v2f*)ap; ap += 4;
          v2f b0 = *(const v2f*)(bt0 + kk);
          v2f b1 = *(const v2f*)(bt1 + kk);
          acc0 = __builtin_amdgcn_wmma_f32_16x16x4_f32(false, a, false, b0, (short)0, acc0, false, false);
          acc1 = __builtin_amdgcn_wmma_f32_16x16x4_f32(false, a, false, b1, (short)0, acc1, false, false);
        }
        __syncthreads();
      }
    }
    epilogue_store(C, bias, acc0, acc1, M, N, ldc, row0, lh, bcol0, bcol1, true, act);
    return;
  }

  // -------- non-LDS path (small / odd-shape GEMMs: K=33 or N=33) ----------
  if (row0 >= M || col0 >= N) return;      // wave-uniform
  const bool hasT1 = (col0 + 16 < N);

  v8f acc0 = {}, acc1 = {};
  if (accum) {
#pragma unroll
    for (int v = 0; v < 8; ++v) {
      int r = row0 + lh * 8 + v;
      if (r < M) {
        if (bcol0 < N)           acc0[v] = C[(long long)r * ldc + bcol0];
        if (hasT1 && bcol1 < N)  acc1[v] = C[(long long)r * ldc + bcol1];
      }
    }
  }

  const bool fast = (row0 + 16 <= M) && (col0 + 32 <= N) && ((K & 3) == 0) &&
                    (((lda | ldb) & 1) == 0);
  if (fast) {
    const float* ap = A + (long long)arow * lda + lh * 2;
    if (transB) {
      const float* bp0 = Bw + (long long)bcol0 * ldb + lh * 2;
      const float* bp1 = Bw + (long long)bcol1 * ldb + lh * 2;
#pragma unroll 4
      for (int k0 = 0; k0 < K; k0 += 4) {
        v2f a  = *(const v2f*)ap;
        v2f b0 = *(const v2f*)bp0;
        v2f b1 = *(const v2f*)bp1;
        acc0 = __builtin_amdgcn_wmma_f32_16x16x4_f32(false, a, false, b0, (short)0, acc0, false, false);
        acc1 = __builtin_amdgcn_wmma_f32_16x16x4_f32(false, a, false, b1, (short)0, acc1, false, false);
        ap += 4; bp0 += 4; bp1 += 4;
      }
    } else {
      const float* bp = Bw + (long long)(lh * 2) * ldb;
#pragma unroll 4
      for (int k0 = 0; k0 < K; k0 += 4) {
        v2f a = *(const v2f*)ap;
        v2f b0, b1;
        b0[0] = bp[bcol0];
        b0[1] = bp[(long long)ldb + bcol0];
        b1[0] = bp[bcol1];
        b1[1] = bp[(long long)ldb + bcol1];
        acc0 = __builtin_amdgcn_wmma_f32_16x16x4_f32(false, a, false, b0, (short)0, acc0, false, false);
        acc1 = __builtin_amdgcn_wmma_f32_16x16x4_f32(false, a, false, b1, (short)0, acc1, false, false);
        ap += 4; bp += (long long)4 * ldb;
      }
    }
  } else {
    for (int k0 = 0; k0 < K; k0 += 4) {
      v2f a, b0, b1;
#pragma unroll
      for (int v = 0; v < 2; ++v) {
        int kk = k0 + lh * 2 + v;
        bool kok = (kk < K);
        a[v] = (kok && arow < M) ? A[(long long)arow * lda + kk] : 0.f;
        float bv0 = 0.f, bv1 = 0.f;
        if (kok) {
          if (bcol0 < N)
            bv0 = transB ? Bw[(long long)bcol0 * ldb + kk] : Bw[(long long)kk * ldb + bcol0];
          if (hasT1 && bcol1 < N)
            bv1 = transB ? Bw[(long long)bcol1 * ldb + kk] : Bw[(long long)kk * ldb + bcol1];
        }
        b0[v] = bv0; b1[v] = bv1;
      }
      acc0 = __builtin_amdgcn_wmma_f32_16x16x4_f32(false, a, false, b0, (short)0, acc0, false, false);
      acc1 = __builtin_amdgcn_wmma_f32_16x16x4_f32(false, a, false, b1, (short)0, acc1, false, false);
    }
  }

  epilogue_store(C, bias, acc0, acc1, M, N, ldc, row0, lh, bcol0, bcol1, hasT1, act);
}

// ---------------------------------------------------------------------------
// RMSNorm over D=512 per row
// ---------------------------------------------------------------------------
__global__ __launch_bounds__(256) void rmsnorm_kernel(
    const float* __restrict__ x, const float* __restrict__ w,
    float* __restrict__ out)
{
  __shared__ float red[256];
  const long long row = blockIdx.x;
  const int t = threadIdx.x;
  float s = 0.f;
  for (int i = t; i < D_; i += 256) { float v = x[row * D_ + i]; s += v * v; }
  red[t] = s; __syncthreads();
  for (int off = 128; off > 0; off >>= 1) {
    if (t < off) red[t] += red[t + off];
    __syncthreads();
  }
  float inv = rsqrtf(red[0] / (float)D_ + 1e-5f);
  for (int i = t; i < D_; i += 256)
    out[row * D_ + i] = x[row * D_ + i] * inv * w[i];
}

// Gaussian mask, length 33: c=17, sigma = mean|i-17| = 273/33
__global__ void gauss33_kernel(float* mask)
{
  if (threadIdx.x == 0 && blockIdx.x == 0) {
    const float c = 17.f, sigma = 273.f / 33.f;
    float w[LH_]; float s = 0.f;
    for (int i = 0; i < LH_; ++i) {
      float d = (float)i - c;
      w[i] = __expf(-0.5f * d * d / (sigma * sigma));
      s += w[i];
    }
    for (int i = 0; i < LH_; ++i) mask[i] = w[i] / s;
  }
}

// x_f = xh[:, :33], x_b = flip(xh[:, 32:]) (SiLU already fused in GEMM)
__global__ void split_fb_kernel(const float* __restrict__ xh,
                                float* __restrict__ xf, float* __restrict__ xb)
{
  long long i = (long long)blockIdx.x * blockDim.x + threadIdx.x;
  if (i >= (long long)B_ * LH_ * DI_) return;
  int d = (int)(i % DI_);
  long long t = i / DI_;
  int l = (int)(t % LH_);
  int b = (int)(t / LH_);
  xf[i] = xh[((long long)b * L_ + l) * DI_ + d];
  xb[i] = xh[((long long)b * L_ + (L_ - 1 - l)) * DI_ + d];
}

// ---------------------------------------------------------------------------
// Selective-scan: per (b,d) thread keeps h[16] in registers, 33 steps.
// ---------------------------------------------------------------------------
__global__ __launch_bounds__(256) void ssm_scan_kernel(
    const float* __restrict__ x, const float* __restrict__ delta,
    const float* __restrict__ dbc, const float* __restrict__ A_log,
    const float* __restrict__ Dp, const float* __restrict__ mask,
    float* __restrict__ ycat, int colOff, int rev)
{
  const int d = blockIdx.x * blockDim.x + threadIdx.x;
  const int b = blockIdx.y;
  if (d >= DI_) return;
  float Aa[NS_], h[NS_];
#pragma unroll
  for (int n = 0; n < NS_; ++n) { Aa[n] = -__expf(A_log[d * NS_ + n]); h[n] = 0.f; }
  const float Dd = Dp[d];
  for (int l = 0; l < LH_; ++l) {
    const long long t = (long long)b * LH_ + l;
    const float dl = delta[t * DI_ + d];
    const float xv = x[t * DI_ + d];
    const float* row = dbc + t * (DTR_ + 2 * NS_);
    float acc = 0.f;
#pragma unroll
    for (int n = 0; n < NS_; ++n) {
      float dA = __expf(dl * Aa[n]);
      h[n] = dA * h[n] + dl * row[DTR_ + n] * xv;     // B_n
      acc += h[n] * row[DTR_ + NS_ + n];              // C_n
    }
    float yv = acc + Dd * xv;
    yv *= mask[rev ? (LH_ - 1 - l) : l];
    ycat[t * (2 * DI_) + colOff + d] = yv;
  }
}

// y[b,l,:] *= mask[l]
__global__ void rowscale_kernel(float* __restrict__ y, const float* __restrict__ mask)
{
  long long i = (long long)blockIdx.x * blockDim.x + threadIdx.x;
  if (i >= (long long)B_ * LH_ * DI_) return;
  int l = (int)((i / DI_) % LH_);
  y[i] *= mask[l];
}

// softmax over last dim (33) of att[b,s,:]
__global__ void softmax33_kernel(float* __restrict__ att)
{
  int r = blockIdx.x * blockDim.x + threadIdx.x;
  if (r >= B_ * L_) return;
  float* p = att + (long long)r * LH_;
  float mx = p[0];
  for (int l = 1; l < LH_; ++l) mx = fmaxf(mx, p[l]);
  float s = 0.f;
  for (int l = 0; l < LH_; ++l) { float e = __expf(p[l] - mx); p[l] = e; s += e; }
  float inv = 1.f / s;
  for (int l = 0; l < LH_; ++l) p[l] *= inv;
}

__global__ void mul_inplace_kernel(float* __restrict__ a,
                                   const float* __restrict__ b, long long n)
{
  long long i = (long long)blockIdx.x * blockDim.x + threadIdx.x;
  if (i < n) a[i] *= b[i];
}

// ---------------------------------------------------------------------------
// Host orchestration
// ---------------------------------------------------------------------------
static inline void gemm(const float* A, const float* Bw, const float* bias, float* C,
                        int M, int N, int K, int lda, int ldb, int ldc,
                        long long sA, long long sB, long long sC,
                        int transB, int accum, int act, int batch, hipStream_t s)
{
  dim3 blk(32, 8, 1);
  dim3 grd((unsigned)((N + 31) / 32), (unsigned)((M + 127) / 128), (unsigned)batch);
  wmma_gemm_f32<<<grd, blk, 0, s>>>(A, Bw, bias, C, M, N, K, lda, ldb, ldc,
                                    sA, sB, sC, transB, accum, act);
}

extern "C" void kernel_launch(void* const* d_in, const int* in_sizes, int n_in,
                              void* d_out, int out_size, void* d_ws, size_t ws_size,
                              hipStream_t stream)
{
  (void)n_in; (void)out_size; (void)ws_size;
  float* ws = (float*)d_ws;
  auto P = [&](int idx, int layer) -> const float* {
    return (const float*)d_in[idx] + (size_t)layer * ((size_t)in_sizes[idx] / NLAYERS_);
  };

  // workspace arena (floats); all offsets are multiples of 32
  constexpr size_t O_MASK = 0;
  constexpr size_t O_NX   = 64;
  constexpr size_t O_XH   = O_NX   + (size_t)B_ * L_  * D_;     // 4160x512
  constexpr size_t O_Z    = O_XH   + (size_t)B_ * L_  * DI_;    // 4160x1024
  constexpr size_t O_XF   = O_Z    + (size_t)B_ * L_  * DI_;
  constexpr size_t O_XB   = O_XF   + (size_t)B_ * LH_ * DI_;
  constexpr size_t O_DBCF = O_XB   + (size_t)B_ * LH_ * DI_;
  constexpr size_t O_DBCB = O_DBCF + (size_t)B_ * LH_ * 64;
  constexpr size_t O_DELF = O_DBCB + (size_t)B_ * LH_ * 64;
  constexpr size_t O_DELB = O_DELF + (size_t)B_ * LH_ * DI_;
  constexpr size_t O_YCAT = O_DELB + (size_t)B_ * LH_ * DI_;
  constexpr size_t O_Y    = O_YCAT + (size_t)B_ * LH_ * 2 * DI_;
  constexpr size_t O_ATT  = O_Y    + (size_t)B_ * LH_ * DI_;
  constexpr size_t O_VV   = O_ATT  + (size_t)B_ * L_  * LH_ + 32;  // 32-float pad
  constexpr size_t O_YTOK = O_VV   + (size_t)B_ * LH_ * DI_;
  constexpr size_t O_LBUF = O_YTOK + (size_t)B_ * L_  * DI_;

  gauss33_kernel<<<dim3(1), dim3(32), 0, stream>>>(ws + O_MASK);

  const int M1 = B_ * L_;    // 4160
  const int M2 = B_ * LH_;   // 2112
  const long long nHalf = (long long)M2 * DI_;
  const long long nFull = (long long)M1 * DI_;

  const float* cur = (const float*)d_in[0];
  for (int layer = 0; layer < NLAYERS_; ++layer) {
    float* outbuf = (layer == NLAYERS_ - 1) ? (float*)d_out : (ws + O_LBUF);

    // 1) RMSNorm
    rmsnorm_kernel<<<dim3((unsigned)M1), dim3(256), 0, stream>>>(cur, P(20, layer), ws + O_NX);
    // 2) xh = silu(nx @ in_x_w^T)   3) z = silu(nx @ in_z_w^T)  (adapool identity)
    gemm(ws + O_NX, P(1, layer), nullptr, ws + O_XH, M1, DI_, D_, D_, D_, DI_, 0, 0, 0, 1, 0, ACT_SILU, 1, stream);
    gemm(ws + O_NX, P(2, layer), nullptr, ws + O_Z,  M1, DI_, D_, D_, D_, DI_, 0, 0, 0, 1, 0, ACT_SILU, 1, stream);
    // 4) split forward / flipped backward halves
    split_fb_kernel<<<dim3((unsigned)((nHalf + 255) / 256)), dim3(256), 0, stream>>>(ws + O_XH, ws + O_XF, ws + O_XB);
    // 5,6) dbc = x @ xproj^T + b   (N = 32 + 16 + 16 = 64)
    gemm(ws + O_XF, P(3, layer), P(4, layer),  ws + O_DBCF, M2, 64, DI_, DI_, DI_, 64, 0, 0, 0, 1, 0, ACT_NONE, 1, stream);
    gemm(ws + O_XB, P(9, layer), P(10, layer), ws + O_DBCB, M2, 64, DI_, DI_, DI_, 64, 0, 0, 0, 1, 0, ACT_NONE, 1, stream);
    // 7,8) delta = softplus(dbc[:, :32] @ dtproj^T + dtb)
    gemm(ws + O_DBCF, P(5, layer),  P(6, layer),  ws + O_DELF, M2, DI_, DTR_, 64, DTR_, DI_, 0, 0, 0, 1, 0, ACT_SOFTPLUS, 1, stream);
    gemm(ws + O_DBCB, P(11, layer), P(12, layer), ws + O_DELB, M2, DI_, DTR_, 64, DTR_, DI_, 0, 0, 0, 1, 0, ACT_SOFTPLUS, 1, stream);
    // 9,10) selective scans -> masked y_f | y_b into y_cat halves
    ssm_scan_kernel<<<dim3(DI_ / 256, B_), dim3(256), 0, stream>>>(
        ws + O_XF, ws + O_DELF, ws + O_DBCF, P(7, layer),  P(8, layer),  ws + O_MASK, ws + O_YCAT, 0,   0);
    ssm_scan_kernel<<<dim3(DI_ / 256, B_), dim3(256), 0, stream>>>(
        ws + O_XB, ws + O_DELB, ws + O_DBCB, P(13, layer), P(14, layer), ws + O_MASK, ws + O_YCAT, DI_, 1);
    // 11) y = y_cat @ pro_to^T + b, then 12) y *= mask[l]
    gemm(ws + O_YCAT, P(15, layer), P(16, layer), ws + O_Y, M2, DI_, 2 * DI_, 2 * DI_, 2 * DI_, DI_, 0, 0, 0, 1, 0, ACT_NONE, 1, stream);
    rowscale_kernel<<<dim3((unsigned)((nHalf + 255) / 256)), dim3(256), 0, stream>>>(ws + O_Y, ws + O_MASK);
    // 13) att[b] = token_wA (65x1024) @ y[b]^T   (batched, transB; N=33 -> non-LDS path)
    gemm(P(17, layer), ws + O_Y, nullptr, ws + O_ATT, L_, LH_, DI_, DI_, DI_, LH_,
         0, (long long)LH_ * DI_, (long long)L_ * LH_, 1, 0, ACT_NONE, B_, stream);
    // 14) softmax over l
    softmax33_kernel<<<dim3((B_ * L_ + 255) / 256), dim3(256), 0, stream>>>(ws + O_ATT);
    // 15) vv[b] = y[b] @ token_wV  (B non-transposed; LDS path)
    gemm(ws + O_Y, P(18, layer), nullptr, ws + O_VV, LH_, DI_, DI_, DI_, DI_, DI_,
         (long long)LH_ * DI_, 0, (long long)LH_ * DI_, 0, 0, ACT_NONE, B_, stream);
    // 16) ytok[b] = att[b] (65x33) @ vv[b] (33x1024)  (K=33 -> non-LDS path)
    gemm(ws + O_ATT, ws + O_VV, nullptr, ws + O_YTOK, L_, DI_, LH_, LH_, DI_, DI_,
         (long long)L_ * LH_, (long long)LH_ * DI_, (long long)L_ * DI_, 0, 0, ACT_NONE, B_, stream);
    // 17) g = ytok * z  (in place)
    mul_inplace_kernel<<<dim3((unsigned)((nFull + 255) / 256)), dim3(256), 0, stream>>>(ws + O_YTOK, ws + O_Z, nFull);
    // 18) mix = g @ out_w^T
    gemm(ws + O_YTOK, P(19, layer), nullptr, outbuf, M1, D_, DI_, DI_, DI_, D_, 0, 0, 0, 1, 0, ACT_NONE, 1, stream);
    // 19) out += xin @ res_pro^T + res_b   (accumulate)
    gemm(cur, P(21, layer), P(22, layer), outbuf, M1, D_, D_, D_, D_, D_, 0, 0, 0, 1, 1, ACT_NONE, 1, stream);

    cur = outbuf;
  }
}